// GatDot_27075473834501
// MI455X (gfx1250) — compile-verified
//
#include <hip/hip_runtime.h>
#include <math.h>

// Problem constants (from reference)
#define BB 512
#define NN 2048
#define DD 128

typedef __attribute__((ext_vector_type(2))) float v2f;
typedef __attribute__((ext_vector_type(8))) float v8f;

// ---------------------------------------------------------------------------
// Kernel 1: Qt = (Q @ W1^T + b1) @ W2  and  c[b] = b2 . (Q@W1^T+b1)[b]
// One wave (32 threads) per 16-row batch tile; 32 blocks total.
// Uses V_WMMA_F32_16X16X4_F32 (exact fp32 matrix path on CDNA5).
//
// Layouts per CDNA5 ISA:
//   A (16x4 f32):  lanes 0-15 -> {K0,K1} of row M=lane; lanes 16-31 -> {K2,K3}
//   B (4x16 f32):  VGPR v: lanes 0-15 = B[K=v][N=lane]; lanes16-31 = B[K=v+2][N=lane-16]
//   C/D (16x16):   VGPR v: lane L holds D[M = v + 8*(L>>4)][N = L&15]
// ---------------------------------------------------------------------------
__global__ void __launch_bounds__(32)
gat_proj_kernel(const float* __restrict__ Q,   // (B, D)
                const float* __restrict__ W1,  // (D, D)
                const float* __restrict__ b1,  // (D)
                const float* __restrict__ W2,  // (D, D)
                const float* __restrict__ b2,  // (D)
                float* __restrict__ Qt,        // (B, D) workspace
                float* __restrict__ cvec)      // (B)    workspace
{
    __shared__ float sQp[16][DD];  // 8 KB: projected Q tile (Qp = Q@W1^T + b1)

    const int m0   = blockIdx.x * 16;   // batch-row tile base
    const int lane = threadIdx.x;       // wave32
    const int l15  = lane & 15;
    const int half = lane >> 4;         // 0 or 1 -> K-group select

    // ---- GEMM 1: Qp[m,d] = sum_e Q[m,e] * W1[d,e] + b1[d] ----
    for (int dt = 0; dt < 8; ++dt) {
        const int d0 = dt * 16;
        v8f acc = {};
        for (int e0 = 0; e0 < DD; e0 += 4) {
            const int kb = e0 + half * 2;
            v2f a, b;
            a.x = Q[(m0 + l15) * DD + kb];
            a.y = Q[(m0 + l15) * DD + kb + 1];
            // B[kk][j] = W1^T[e0+kk][d0+j] = W1[(d0+j)*D + e0+kk]
            b.x = W1[(d0 + l15) * DD + kb];
            b.y = W1[(d0 + l15) * DD + kb + 1];
            acc = __builtin_amdgcn_wmma_f32_16x16x4_f32(
                false, a, false, b, (short)0, acc, false, false);
        }
        const float bias = b1[d0 + l15];
        #pragma unroll
        for (int v = 0; v < 8; ++v)
            sQp[v + 8 * half][d0 + l15] = acc[v] + bias;
    }
    __syncthreads();  // single-wave WG: LDS write->read ordering

    // ---- GEMM 2: Qt[m,e'] = sum_d Qp[m,d] * W2[d,e'] ----
    for (int et = 0; et < 8; ++et) {
        const int e0 = et * 16;
        v8f acc = {};
        for (int d0 = 0; d0 < DD; d0 += 4) {
            const int kb = d0 + half * 2;
            v2f a, b;
            a.x = sQp[l15][kb];
            a.y = sQp[l15][kb + 1];
            // B[kk][j] = W2[(d0+kk)][e0+j]
            b.x = W2[(kb)     * DD + e0 + l15];
            b.y = W2[(kb + 1) * DD + e0 + l15];
            acc = __builtin_amdgcn_wmma_f32_16x16x4_f32(
                false, a, false, b, (short)0, acc, false, false);
        }
        #pragma unroll
        for (int v = 0; v < 8; ++v)
            Qt[(size_t)(m0 + v + 8 * half) * DD + e0 + l15] = acc[v];
    }

    // ---- c[b] = b2 . Qp[b,:]  (cancels in softmax; kept for fidelity) ----
    if (lane < 16) {
        float s = 0.f;
        for (int d = 0; d < DD; ++d) s += b2[d] * sQp[lane][d];
        cvec[m0 + lane] = s;
    }
}

// ---------------------------------------------------------------------------
// Kernel 2: per-batch streaming attention (bandwidth-bound: K + V read once).
// 512 blocks (one per batch), 256 threads = 8 waves.
//   pass 1: alpha[n] = K[b,n,:].Qt[b] + c[b] - (1-adj)*1e30 ; masked softmax
//   pass 2: attn_sum[d] = sum_n w[n] * V[b,n,d]
// Each wave handles one K/V row per iteration: lane l loads float4 at d=4l,
// so one wavefront load = one full 512B row, perfectly coalesced.
// ---------------------------------------------------------------------------
__global__ void __launch_bounds__(256)
gat_attn_kernel(const float* __restrict__ K,    // (B, N, D)
                const float* __restrict__ V,    // (B, N, D)
                const float* __restrict__ adj,  // (B, N)
                const float* __restrict__ Qt,   // (B, D)
                const float* __restrict__ cvec, // (B)
                float* __restrict__ out_w,      // (B, 1, N)
                float* __restrict__ out_sum)    // (B, D)
{
    __shared__ float sAlpha[NN];   // 8 KB
    __shared__ float sRed[8];
    __shared__ float sAcc[DD];

    const int b    = blockIdx.x;
    const int tid  = threadIdx.x;
    const int wave = tid >> 5;
    const int lane = tid & 31;

    // Qt[b] distributed across lanes: lane owns dims [4*lane, 4*lane+4)
    const float4 qt = ((const float4*)(Qt + (size_t)b * DD))[lane];
    const float  cb = cvec[b];

    // ---- pass 1: logits ----
    const float* Kb = K + (size_t)b * NN * DD;
    for (int n = wave; n < NN; n += 8) {
        const float4 kv = ((const float4*)(Kb + (size_t)n * DD))[lane];
        float p = kv.x * qt.x + kv.y * qt.y + kv.z * qt.z + kv.w * qt.w;
        #pragma unroll
        for (int off = 16; off >= 1; off >>= 1)
            p += __shfl_xor(p, off, 32);
        if (lane == 0) sAlpha[n] = p;
    }
    __syncthreads();

    // ---- mask + row max ----
    float lmax = -INFINITY;
    for (int n = tid; n < NN; n += 256) {
        float a = sAlpha[n] + cb - (1.0f - adj[(size_t)b * NN + n]) * 1e30f;
        sAlpha[n] = a;
        lmax = fmaxf(lmax, a);
    }
    #pragma unroll
    for (int off = 16; off >= 1; off >>= 1)
        lmax = fmaxf(lmax, __shfl_xor(lmax, off, 32));
    if (lane == 0) sRed[wave] = lmax;
    __syncthreads();
    float bmax = sRed[0];
    #pragma unroll
    for (int w = 1; w < 8; ++w) bmax = fmaxf(bmax, sRed[w]);

    // ---- exp + sum ----
    float lsum = 0.f;
    for (int n = tid; n < NN; n += 256) {
        const float e = __expf(sAlpha[n] - bmax);
        sAlpha[n] = e;
        lsum += e;
    }
    #pragma unroll
    for (int off = 16; off >= 1; off >>= 1)
        lsum += __shfl_xor(lsum, off, 32);
    __syncthreads();                 // all reads of sRed (max) done before reuse
    if (lane == 0) sRed[wave] = lsum;
    __syncthreads();
    float bsum = 0.f;
    #pragma unroll
    for (int w = 0; w < 8; ++w) bsum += sRed[w];
    const float inv = 1.0f / bsum;

    // ---- normalize + emit attention weights ----
    for (int n = tid; n < NN; n += 256) {
        const float w = sAlpha[n] * inv;
        sAlpha[n] = w;
        out_w[(size_t)b * NN + n] = w;
    }
    if (tid < DD) sAcc[tid] = 0.f;
    __syncthreads();

    // ---- pass 2: weighted sum over V ----
    const float* Vb = V + (size_t)b * NN * DD;
    float4 acc = make_float4(0.f, 0.f, 0.f, 0.f);
    for (int n = wave; n < NN; n += 8) {
        const float  w  = sAlpha[n];
        const float4 vv = ((const float4*)(Vb + (size_t)n * DD))[lane];
        acc.x += w * vv.x;  acc.y += w * vv.y;
        acc.z += w * vv.z;  acc.w += w * vv.w;
    }
    // cross-wave reduction via LDS float atomics (ds_add_f32)
    atomicAdd(&sAcc[lane * 4 + 0], acc.x);
    atomicAdd(&sAcc[lane * 4 + 1], acc.y);
    atomicAdd(&sAcc[lane * 4 + 2], acc.z);
    atomicAdd(&sAcc[lane * 4 + 3], acc.w);
    __syncthreads();
    if (tid < DD) out_sum[(size_t)b * DD + tid] = sAcc[tid];
}

// ---------------------------------------------------------------------------
// Inputs (setup_inputs order): 0:Q 1:K 2:V 3:adj 4:s_mask 5:W1 6:b1 7:W2 8:b2
// Output: attn_weight (B*1*N floats) then attn_sum (B*D floats), concatenated.
// ---------------------------------------------------------------------------
extern "C" void kernel_launch(void* const* d_in, const int* in_sizes, int n_in,
                              void* d_out, int out_size, void* d_ws, size_t ws_size,
                              hipStream_t stream) {
    const float* Q   = (const float*)d_in[0];
    const float* K   = (const float*)d_in[1];
    const float* V   = (const float*)d_in[2];
    const float* adj = (const float*)d_in[3];
    const float* W1  = (const float*)d_in[5];
    const float* b1  = (const float*)d_in[6];
    const float* W2  = (const float*)d_in[7];
    const float* b2  = (const float*)d_in[8];

    float* Qt   = (float*)d_ws;                 // B*D floats
    float* cvec = Qt + (size_t)BB * DD;         // B floats

    float* out_w   = (float*)d_out;             // B*N
    float* out_sum = out_w + (size_t)BB * NN;   // B*D

    gat_proj_kernel<<<BB / 16, 32, 0, stream>>>(Q, W1, b1, W2, b2, Qt, cvec);
    gat_attn_kernel<<<BB, 256, 0, stream>>>(K, V, adj, Qt, cvec, out_w, out_sum);
}